// NaiveBias_36876589204102
// MI455X (gfx1250) — compile-verified
//
#include <hip/hip_runtime.h>

typedef float v2f __attribute__((ext_vector_type(2)));
typedef float v8f __attribute__((ext_vector_type(8)));

#define Bc     4
#define LFULL  2050
#define Lq     2048
#define Hh     16
#define Dd     64
#define SHAPE  4095
#define WSTR   4096          // wexp row stride in workspace (tail element zeroed)

#define KT     32            // K-slab per iteration
#define MT     256           // M rows per workgroup (8 waves x 32 rows)
#define BAND   (KT + MT)     // 288 wexp values cover the whole Toeplitz tile
#define VSTR   (KT + 2)      // padded LDS stride for transposed V slab

// ---------------------------------------------------------------------------
// Kernel 1: wexp = exp(w - offset), prefix-sum -> z_pb, zero pbv padding rows.
// ---------------------------------------------------------------------------
__global__ __launch_bounds__(256) void prep_kernel(const float* __restrict__ w,
                                                   const float* __restrict__ offset,
                                                   float* __restrict__ wexp,
                                                   float* __restrict__ out)
{
    __shared__ float s[4096];
    __shared__ float part[256];
    const int h   = blockIdx.x;
    const int tid = threadIdx.x;
    const float off = offset[h];

    for (int k = tid; k < 4096; k += 256) {
        float val = 0.f;
        if (k < SHAPE) val = __expf(w[h * SHAPE + k] - off);
        wexp[h * WSTR + k] = val;      // tail (k==4095) zeroed: band staging may read it
        s[k] = val;
    }
    __syncthreads();

    // block-wide exclusive prefix sum over 4096 elements (16 per thread)
    const int base = tid * 16;
    float loc[16];
    float acc = 0.f;
    #pragma unroll
    for (int i = 0; i < 16; ++i) { loc[i] = acc; acc += s[base + i]; }
    part[tid] = acc;
    __syncthreads();
    if (tid == 0) {
        float run = 0.f;
        for (int i = 0; i < 256; ++i) { float t = part[i]; part[i] = run; run += t; }
    }
    __syncthreads();
    const float pbase = part[tid];
    #pragma unroll
    for (int i = 0; i < 16; ++i) s[base + i] = pbase + loc[i];   // S_excl[t]
    __syncthreads();

    // z_core[h,i] = S[i+2048] - S[i]; scatter to z_pb (1,B,L+2,H) with padding
    float* zout = out + (size_t)Bc * LFULL * Hh * Dd;
    for (int i = tid; i < Lq; i += 256) {
        const float z = s[i + Lq] - s[i];
        #pragma unroll
        for (int n = 0; n < Bc; ++n)
            zout[((size_t)n * LFULL + (i + 1)) * Hh + h] = z;
    }
    if (tid < 2 * Bc) {
        const int n  = tid & 3;
        const int jp = (tid >> 2) ? (LFULL - 1) : 0;
        zout[((size_t)n * LFULL + jp) * Hh + h] = 0.f;
    }
    // zero pbv padding rows j=0 and j=L+1 for this head
    for (int i = tid; i < Bc * 2 * Dd; i += 256) {
        const int d  = i & 63;
        const int r  = i >> 6;
        const int n  = r & 3;
        const int jp = (r >> 2) ? (LFULL - 1) : 0;
        out[(((size_t)n * LFULL + jp) * Hh + h) * Dd + d] = 0.f;
    }
}

// ---------------------------------------------------------------------------
// Kernel 2: Toeplitz GEMM per head via V_WMMA_F32_16X16X4_F32.
//   Out[j, d] = sum_l wexp[h, 2047 + l - j] * V[n, l+1, h, d]
// Workgroup: 256 rows (j) x 64 cols (d) for one (n, h). 8 waves, each 32x64.
// ---------------------------------------------------------------------------
__global__ __launch_bounds__(256) void pbv_kernel(const float* __restrict__ v,
                                                  const float* __restrict__ wexp,
                                                  float* __restrict__ out)
{
    __shared__ float band[BAND];
    __shared__ float vlds[Dd * VSTR];

    const int tid  = threadIdx.x;
    const int lane = tid & 31;
    const int wave = tid >> 5;            // 0..7
    const int m    = lane & 15;
    const int kh   = lane >> 4;           // 0/1 -> K pairs {0,1}/{2,3}
    const int j0   = blockIdx.x * MT;     // 0..1792
    const int n    = blockIdx.y;
    const int h    = blockIdx.z;

    const float* wrow = wexp + h * WSTR;
    const int bandBase = (Lq - 1) - j0 - (MT - 1);   // + l0 added per K-step

    v8f c[2][4] = {};                     // 2 M-frags x 4 N-frags, 16x16 each

    // V row l -> v[((n*LFULL + l + 1)*H + h)*D + d]
    const float* vbase = v + (((size_t)n * LFULL + 1) * Hh + h) * Dd;
    const int vrowk = tid >> 3;           // kk within slab, 0..31
    const int vd    = (tid & 7) * 8;      // d base, 0..56

    for (int l0 = 0; l0 < Lq; l0 += KT) {
        // ---- stage Toeplitz band (288 floats) ----
        band[tid] = wrow[bandBase + l0 + tid];
        if (tid < BAND - 256) band[256 + tid] = wrow[bandBase + l0 + 256 + tid];

        // ---- stage V slab transposed: vlds[d*VSTR + kk] = V[l0+kk, d] ----
        {
            const float* src = vbase + (size_t)(l0 + vrowk) * (Hh * Dd) + vd;
            const float4 f0 = *(const float4*)(src);
            const float4 f1 = *(const float4*)(src + 4);
            vlds[(vd + 0) * VSTR + vrowk] = f0.x;
            vlds[(vd + 1) * VSTR + vrowk] = f0.y;
            vlds[(vd + 2) * VSTR + vrowk] = f0.z;
            vlds[(vd + 3) * VSTR + vrowk] = f0.w;
            vlds[(vd + 4) * VSTR + vrowk] = f1.x;
            vlds[(vd + 5) * VSTR + vrowk] = f1.y;
            vlds[(vd + 6) * VSTR + vrowk] = f1.z;
            vlds[(vd + 7) * VSTR + vrowk] = f1.w;
        }
        __syncthreads();

        const int wr = wave * 32;
        #pragma unroll
        for (int ks = 0; ks < KT / 4; ++ks) {
            // A fragments (16x4 f32 layout: lane m, VGPR0=K{0,2}, VGPR1=K{1,3})
            v2f a[2];
            #pragma unroll
            for (int f = 0; f < 2; ++f) {
                const int t0 = ks * 4 + 2 * kh + (MT - 1) - (wr + f * 16 + m);
                a[f][0] = band[t0];
                a[f][1] = band[t0 + 1];
            }
            // B fragments: columns d = nt*16 + m, contiguous K pair in vlds
            #pragma unroll
            for (int nt = 0; nt < 4; ++nt) {
                const float* bp = &vlds[(nt * 16 + m) * VSTR + ks * 4 + 2 * kh];
                v2f b;
                b[0] = bp[0];
                b[1] = bp[1];
                #pragma unroll
                for (int f = 0; f < 2; ++f) {
                    c[f][nt] = __builtin_amdgcn_wmma_f32_16x16x4_f32(
                        false, a[f], false, b, (short)0, c[f][nt], false, false);
                }
            }
        }
        __syncthreads();
    }

    // ---- write out: pbv[n, j+1, h, d]; C layout: VGPR r -> M = r (+8 hi half) ----
    #pragma unroll
    for (int f = 0; f < 2; ++f) {
        #pragma unroll
        for (int nt = 0; nt < 4; ++nt) {
            const int d = nt * 16 + m;
            #pragma unroll
            for (int r = 0; r < 8; ++r) {
                const int mrow = (lane < 16) ? r : (r + 8);
                const int j = j0 + wave * 32 + f * 16 + mrow;
                out[(((size_t)n * LFULL + (j + 1)) * Hh + h) * Dd + d] = c[f][nt][r];
            }
        }
    }
}

extern "C" void kernel_launch(void* const* d_in, const int* in_sizes, int n_in,
                              void* d_out, int out_size, void* d_ws, size_t ws_size,
                              hipStream_t stream)
{
    (void)in_sizes; (void)n_in; (void)out_size; (void)ws_size;
    const float* v      = (const float*)d_in[0];
    const float* offset = (const float*)d_in[1];
    const float* w      = (const float*)d_in[2];
    float*       out    = (float*)d_out;
    float*       wexp   = (float*)d_ws;          // 16 * 4096 * 4 B = 256 KB

    prep_kernel<<<dim3(Hh), 256, 0, stream>>>(w, offset, wexp, out);
    pbv_kernel<<<dim3(Lq / MT, Bc, Hh), 256, 0, stream>>>(v, wexp, out);
}